// ConvOP_9723805958411
// MI455X (gfx1250) — compile-verified
//
#include <hip/hip_runtime.h>
#include <hip/hip_bf16.h>

typedef __attribute__((ext_vector_type(2))) float v2f;
typedef __attribute__((ext_vector_type(8))) float v8f;

#define KPTS   15
#define HNB    32
#define CIN    64
#define COUT   128
#define KPEXT  1.0f
#define KDIM   (KPTS * CIN)           // 960 contraction depth for phase 2

// LDS layout (floats), padded strides chosen for bank-conflict-free access
#define QTILE    32                   // queries per block (2 WMMA M-tiles)
#define WF_STR   964                  // 960 (k*64+c) + pad 4 -> 4*m mod 64 distinct banks
#define WST_STR  34                   // 16 k-rows x 32 h, pad 2 -> 34*m mod 64 distinct
#define WF_FLOATS   (QTILE * WF_STR)          // 30848
#define WST_FLOATS  (16 * WST_STR)            // 544 per wave
#define SMEM_BYTES  ((WF_FLOATS + 8 * WST_FLOATS) * 4)   // 140800 B (< 160KB)

// ---- one-shot weight repack: W[960][128] -> Wpk[pair p][o*2 + {0,1}] ----
// so a B fragment (rows kb,kb+1 at column o) is one coalesced b64 load.
__global__ __launch_bounds__(256)
void repack_weights(const float* __restrict__ wts, float* __restrict__ wpk)
{
    const int tid = blockIdx.x * 256 + threadIdx.x;     // [0, 480*128)
    const int p = tid >> 7;                              // row pair 0..479
    const int o = tid & 127;                             // output column
    v2f v;
    v.x = wts[(size_t)(2 * p) * COUT + o];
    v.y = wts[(size_t)(2 * p + 1) * COUT + o];
    *(v2f*)(wpk + (size_t)p * (2 * COUT) + 2 * o) = v;
}

__global__ __launch_bounds__(256)
void kpconv_wmma_f32(const float* __restrict__ q_pts,
                     const float* __restrict__ s_pts,
                     const int*   __restrict__ nbi,
                     const float* __restrict__ x,
                     const float* __restrict__ wts,   // [960, COUT] flat
                     const float* __restrict__ wpk,   // packed pairs (may be null)
                     const float* __restrict__ kp,    // [K, 3]
                     float* __restrict__ out,
                     int N, int M, int usePk)
{
    extern __shared__ float smem[];
    float* wf  = smem;                                           // [QTILE][WF_STR]
    float* wst = smem + WF_FLOATS + (threadIdx.x >> 5) * WST_FLOATS; // per-wave [16][WST_STR]

    const int lane = threadIdx.x & 31;
    const int wave = threadIdx.x >> 5;
    const int m    = lane & 15;                  // tile row/col selector
    const int lo   = lane >> 4;                  // lane-half -> K subgroup
    const int n0   = blockIdx.x * QTILE;

    // ------------- Phase 1: wf[r][k*64+c] = sum_h w[k][h] * x[idx[h]][c] -------------
    for (int rr = 0; rr < QTILE / 8; ++rr) {     // 4 queries per wave
        const int r  = wave * (QTILE / 8) + rr;
        const int n  = n0 + r;
        const int nc = (n < N) ? n : (N - 1);

        const int h   = lane;                    // lane <-> neighbor
        int idx       = nbi[(size_t)nc * HNB + h];
        const bool ok = (idx >= 0) && (idx < M);
        const int ic  = ok ? idx : 0;

        const float qx = q_pts[nc * 3 + 0];
        const float qy = q_pts[nc * 3 + 1];
        const float qz = q_pts[nc * 3 + 2];
        const float px = (ok ? s_pts[(size_t)ic * 3 + 0] : 1.0e6f) - qx;
        const float py = (ok ? s_pts[(size_t)ic * 3 + 1] : 1.0e6f) - qy;
        const float pz = (ok ? s_pts[(size_t)ic * 3 + 2] : 1.0e6f) - qz;

#pragma unroll
        for (int k = 0; k < KPTS; ++k) {
            const float dx = px - kp[k * 3 + 0];
            const float dy = py - kp[k * 3 + 1];
            const float dz = pz - kp[k * 3 + 2];
            float w = 1.0f - sqrtf(dx * dx + dy * dy + dz * dz) / KPEXT;
            w = (w > 0.0f) ? w : 0.0f;
            wst[k * WST_STR + h] = w;            // transposed: [k][h]
        }
        wst[15 * WST_STR + h] = 0.0f;            // zero pad row (K 15 -> 16)
        __builtin_amdgcn_wave_barrier();         // LDS is in-order per wave

        // D[16x16]: rows=k, cols=c-subtile; contraction over h (32) in 8 steps of 4
#pragma unroll
        for (int ct = 0; ct < 4; ++ct) {
            v8f acc = {};
#pragma unroll
            for (int kk = 0; kk < 8; ++kk) {
                const int hh = kk * 4 + lo * 2;
                v2f a = *(const v2f*)(wst + m * WST_STR + hh);   // w[m][hh..hh+1]
                const int r0 = __shfl(ic, hh);
                const int r1 = __shfl(ic, hh + 1);
                v2f b;
                b.x = x[(size_t)r0 * CIN + ct * 16 + m];
                b.y = x[(size_t)r1 * CIN + ct * 16 + m];
                acc = __builtin_amdgcn_wmma_f32_16x16x4_f32(
                          false, a, false, b, (short)0, acc, false, false);
            }
#pragma unroll
            for (int j = 0; j < 8; ++j) {
                const int kr = j + lo * 8;
                if (kr < KPTS)
                    wf[r * WF_STR + kr * CIN + ct * 16 + m] = acc[j];
            }
        }
        __builtin_amdgcn_wave_barrier();
    }
    __syncthreads();

    // ------------- Phase 2: out[QTILE x COUT] = wf[QTILE x 960] x W[960 x COUT] -------------
    // Each wave owns 16 output columns; two M-tiles (rows 0-15, 16-31) share every B fragment.
    {
        const int o = wave * 16 + m;
        v8f acc0 = {};
        v8f acc1 = {};
        if (usePk) {
            for (int kc = 0; kc < KDIM / 4; ++kc) {
                const int kb = kc * 4 + lo * 2;
                v2f a0 = *(const v2f*)(wf + m * WF_STR + kb);
                v2f a1 = *(const v2f*)(wf + (16 + m) * WF_STR + kb);
                v2f b  = *(const v2f*)(wpk + (size_t)(kb >> 1) * (2 * COUT) + 2 * o);
                acc0 = __builtin_amdgcn_wmma_f32_16x16x4_f32(
                           false, a0, false, b, (short)0, acc0, false, false);
                acc1 = __builtin_amdgcn_wmma_f32_16x16x4_f32(
                           false, a1, false, b, (short)0, acc1, false, false);
            }
        } else {
            for (int kc = 0; kc < KDIM / 4; ++kc) {
                const int kb = kc * 4 + lo * 2;
                v2f a0 = *(const v2f*)(wf + m * WF_STR + kb);
                v2f a1 = *(const v2f*)(wf + (16 + m) * WF_STR + kb);
                v2f b;
                b.x = wts[(size_t)kb * COUT + o];
                b.y = wts[(size_t)(kb + 1) * COUT + o];
                acc0 = __builtin_amdgcn_wmma_f32_16x16x4_f32(
                           false, a0, false, b, (short)0, acc0, false, false);
                acc1 = __builtin_amdgcn_wmma_f32_16x16x4_f32(
                           false, a1, false, b, (short)0, acc1, false, false);
            }
        }
#pragma unroll
        for (int j = 0; j < 8; ++j) {
            const int q = j + lo * 8;
            const int nood = n0 + q;
            if (nood < N) out[(size_t)nood * COUT + o] = acc0[j];
            const int n1 = n0 + 16 + q;
            if (n1 < N)  out[(size_t)n1 * COUT + o] = acc1[j];
        }
    }
}

extern "C" void kernel_launch(void* const* d_in, const int* in_sizes, int n_in,
                              void* d_out, int out_size, void* d_ws, size_t ws_size,
                              hipStream_t stream)
{
    const float* q_pts = (const float*)d_in[0];   // [N,3]
    const float* s_pts = (const float*)d_in[1];   // [M,3]
    const int*   nbi   = (const int*)d_in[2];     // [N,H]
    const float* x     = (const float*)d_in[3];   // [M,CIN]
    const float* wts   = (const float*)d_in[4];   // [K,CIN,COUT] == [960,COUT]
    const float* kp    = (const float*)d_in[5];   // [K,3]
    float* out = (float*)d_out;

    const int N = in_sizes[0] / 3;
    const int M = in_sizes[1] / 3;

    const size_t pk_bytes = (size_t)KDIM * COUT * sizeof(float);   // 491520
    const int usePk = (ws_size >= pk_bytes) ? 1 : 0;
    float* wpk = (float*)d_ws;

    if (usePk) {
        // 480 row-pairs x 128 cols = 61440 threads
        repack_weights<<<(KDIM / 2) * COUT / 256, 256, 0, stream>>>(wts, wpk);
    }

    const int blocks = (N + QTILE - 1) / QTILE;
    kpconv_wmma_f32<<<blocks, 256, SMEM_BYTES, stream>>>(
        q_pts, s_pts, nbi, x, wts, wpk, kp, out, N, M, usePk);
}